// ProjectToPlane_21887153341038
// MI455X (gfx1250) — compile-verified
//
#include <hip/hip_runtime.h>
#include <math.h>

typedef __attribute__((ext_vector_type(2))) float v2f;
typedef __attribute__((ext_vector_type(8))) float v8f;

#define NPTS   8192
#define NTILES (NPTS / 16)
#define GRID_W 96

// ---------------------------------------------------------------------------
// Kernel 0: precompute squared norms and WMMA-lane-layout operand array.
// W[tile*32 + lane]: lane<16 -> (x,y) of point tile*16+lane    (K=0,1)
//                    lane>=16 -> (z,0) of point tile*16+lane-16 (K=2,3 pad)
// ---------------------------------------------------------------------------
__global__ __launch_bounds__(256) void prep_kernel(const float* __restrict__ pts,
                                                   float* __restrict__ sq,
                                                   v2f* __restrict__ W)
{
    const int p = blockIdx.x * 256 + threadIdx.x;   // 32 blocks * 256 = 8192
    const float x = pts[p * 3 + 0] * 96.0f;
    const float y = pts[p * 3 + 1] * 96.0f;
    const float z = pts[p * 3 + 2] * 96.0f;
    sq[p] = x * x + y * y + z * z;
    const int tile = p >> 4, m = p & 15;
    v2f lo; lo.x = x; lo.y = y;
    v2f hi; hi.x = z; hi.y = 0.0f;
    W[tile * 32 + m]      = lo;
    W[tile * 32 + 16 + m] = hi;
}

// ---------------------------------------------------------------------------
// Kernel 1: nearest-neighbor min distance via WMMA f32 16x16x4 Gram tiles.
// One wave per 16-row block. tile index pinned to an SGPR and the J loop is
// split into [0,tile) and (tile,NTILES) so the hot loop has no per-iteration
// guard and EXEC is never touched around the WMMA.
// D layout: lane l holds D[v + 8*(l>=16)][l&15] in g[v].
// ---------------------------------------------------------------------------
__global__ __launch_bounds__(256) void nn_kernel(const float* __restrict__ sq,
                                                 const v2f* __restrict__ W,
                                                 float* __restrict__ nn)
{
    const int lane    = threadIdx.x & 31;
    const int wave    = threadIdx.x >> 5;
    // wave-uniform by construction; make it SGPR so loop bounds are scalar
    const int tile    = __builtin_amdgcn_readfirstlane(blockIdx.x * 8 + wave);
    const int I       = tile * 16;
    const int m       = lane & 15;
    const bool hiHalf = (lane >= 16);
    const int M_base  = hiHalf ? 8 : 0;

    const v2f a = W[tile * 32 + lane];

    float sqRow[8];
#pragma unroll
    for (int v = 0; v < 8; ++v) sqRow[v] = sq[I + M_base + v];

    float minv[8];
#pragma unroll
    for (int v = 0; v < 8; ++v) minv[v] = INFINITY;

    // off-diagonal tiles below the diagonal tile
    for (int jt = 0; jt < tile; ++jt) {
        const v2f   b   = W[jt * 32 + lane];
        const float sqj = sq[jt * 16 + m];
        v8f c = {};
        v8f g = __builtin_amdgcn_wmma_f32_16x16x4_f32(
            false, a, false, b, (short)0, c, false, false);
#pragma unroll
        for (int v = 0; v < 8; ++v)
            minv[v] = fminf(minv[v], sqRow[v] + sqj - 2.0f * g[v]);
    }
    // off-diagonal tiles above the diagonal tile
    for (int jt = tile + 1; jt < NTILES; ++jt) {
        const v2f   b   = W[jt * 32 + lane];
        const float sqj = sq[jt * 16 + m];
        v8f c = {};
        v8f g = __builtin_amdgcn_wmma_f32_16x16x4_f32(
            false, a, false, b, (short)0, c, false, false);
#pragma unroll
        for (int v = 0; v < 8; ++v)
            minv[v] = fminf(minv[v], sqRow[v] + sqj - 2.0f * g[v]);
    }

    // peeled diagonal tile: exclude gi == gj elements
    {
        const float sqj = sq[I + m];
        v8f c = {};
        v8f g = __builtin_amdgcn_wmma_f32_16x16x4_f32(
            false, a, false, a, (short)0, c, false, false);
#pragma unroll
        for (int v = 0; v < 8; ++v) {
            const float d2 = sqRow[v] + sqj - 2.0f * g[v];
            if (M_base + v != m) minv[v] = fminf(minv[v], d2);
        }
    }

    // min across the 16 lanes of each half (xor masks stay within halves)
#pragma unroll
    for (int off = 1; off < 16; off <<= 1) {
#pragma unroll
        for (int v = 0; v < 8; ++v)
            minv[v] = fminf(minv[v], __shfl_xor(minv[v], off, 32));
    }

    if (m == 0) {
#pragma unroll
        for (int v = 0; v < 8; ++v)
            nn[I + M_base + v] = sqrtf(fmaxf(minv[v], 0.0f));
    }
}

// ---------------------------------------------------------------------------
// Kernel 2: delta2 = M * (M * mean(nn)) = 9 * sum(nn) / 8192
// ---------------------------------------------------------------------------
__global__ __launch_bounds__(256) void reduce_kernel(const float* __restrict__ nn,
                                                     float* __restrict__ delta)
{
    __shared__ float s[256];
    float t = 0.0f;
    for (int i = threadIdx.x; i < NPTS; i += 256) t += nn[i];
    s[threadIdx.x] = t;
    __syncthreads();
    for (int o = 128; o > 0; o >>= 1) {
        if (threadIdx.x < (unsigned)o) s[threadIdx.x] += s[threadIdx.x + o];
        __syncthreads();
    }
    if (threadIdx.x == 0) delta[0] = 9.0f * s[0] / 8192.0f;
}

// ---------------------------------------------------------------------------
// Kernel 3: per-pixel masked median via exact 4-round radix select.
// Raw points staged ONCE into LDS with global_load_async_to_lds_b128
// (ASYNCcnt-tracked), then all 4 passes run from LDS.
// ---------------------------------------------------------------------------
__device__ __forceinline__ unsigned f2u(float f) {
    unsigned u = __float_as_uint(f);
    return (u & 0x80000000u) ? ~u : (u | 0x80000000u);
}
__device__ __forceinline__ float u2f(unsigned u) {
    u = (u & 0x80000000u) ? (u & 0x7FFFFFFFu) : ~u;
    return __uint_as_float(u);
}

__global__ __launch_bounds__(256) void project_kernel(const float* __restrict__ pts,
                                                      const float* __restrict__ delta_p,
                                                      float* __restrict__ out)
{
    __shared__ unsigned h0[256];
    __shared__ unsigned h1[256];
    __shared__ unsigned s_prefix[2];
    __shared__ unsigned s_rank[2];
    __shared__ int s_count;
    extern __shared__ float s_raw[];               // NPTS*3 floats = 96 KB

    const int tid = threadIdx.x;

    // ---- async stage: 96 KB of raw points -> LDS (6144 x b128, 24/thread) ----
    {
        const unsigned ldsoff = (unsigned)(size_t)(s_raw);
        const char* gp = (const char*)pts;
        for (int t = tid; t < (NPTS * 12) / 16; t += 256) {
            const unsigned laddr = ldsoff + (unsigned)t * 16u;
            const unsigned long long ga = (unsigned long long)(gp + (size_t)t * 16u);
            asm volatile("global_load_async_to_lds_b128 %0, %1, off"
                         :: "v"(laddr), "v"(ga) : "memory");
        }
        asm volatile("s_wait_asynccnt 0x0" ::: "memory");
    }
    __syncthreads();

    const int p  = blockIdx.x;
    const int iy = p / GRID_W;
    const int ix = p % GRID_W;
    const float Xr = (float)(ix - 48);
    const float Yr = (float)(iy - 48);
    const float d  = delta_p[0];
    // Replicate reference exactly, including px upper bound using Yr:
    const float xlo = Xr - d, xhi = Yr + d;
    const float ylo = Yr - d, yhi = Yr + d;

    // --- round 0: count + top-byte histogram ---
    h0[tid] = 0u; h1[tid] = 0u;
    __syncthreads();
    for (int i = tid; i < NPTS; i += 256) {
        const float px = s_raw[i * 3 + 0] * 96.0f;
        const float py = s_raw[i * 3 + 1] * 96.0f;
        if (px >= xlo && px <= xhi && py >= ylo && py <= yhi) {
            const float pz = s_raw[i * 3 + 2] * 96.0f;
            atomicAdd(&h0[f2u(pz) >> 24], 1u);
        }
    }
    __syncthreads();
    if (tid == 0) {
        unsigned c = 0;
        for (int b = 0; b < 256; ++b) c += h0[b];
        s_count = (int)c;
    }
    __syncthreads();
    const int c = s_count;
    if (c == 0) {
        if (tid == 0) out[p] = 0.0f;
        return;
    }
    if (tid < 2) {
        unsigned k = (tid == 0) ? (unsigned)((c - 1) >> 1) : (unsigned)(c >> 1);
        unsigned cum = 0, pref = 0;
        for (int b = 0; b < 256; ++b) {
            const unsigned hh = h0[b];
            if (k < cum + hh) { pref = (unsigned)b << 24; k -= cum; break; }
            cum += hh;
        }
        s_prefix[tid] = pref;
        s_rank[tid]   = k;
    }
    __syncthreads();

    // --- rounds 1..3: dual-rank refinement ---
    for (int r = 1; r < 4; ++r) {
        const int shift = 24 - 8 * r;
        const unsigned maskHigh = 0xFFFFFFFFu << (shift + 8);
        h0[tid] = 0u; h1[tid] = 0u;
        __syncthreads();
        const unsigned pre0 = s_prefix[0];
        const unsigned pre1 = s_prefix[1];
        for (int i = tid; i < NPTS; i += 256) {
            const float px = s_raw[i * 3 + 0] * 96.0f;
            const float py = s_raw[i * 3 + 1] * 96.0f;
            if (px >= xlo && px <= xhi && py >= ylo && py <= yhi) {
                const float pz = s_raw[i * 3 + 2] * 96.0f;
                const unsigned bits = f2u(pz);
                const unsigned byte = (bits >> shift) & 0xFFu;
                if ((bits & maskHigh) == pre0) atomicAdd(&h0[byte], 1u);
                if ((bits & maskHigh) == pre1) atomicAdd(&h1[byte], 1u);
            }
        }
        __syncthreads();
        if (tid < 2) {
            unsigned* h = (tid == 0) ? h0 : h1;
            unsigned k = s_rank[tid], cum = 0;
            unsigned pref = s_prefix[tid];
            for (int b = 0; b < 256; ++b) {
                const unsigned hh = h[b];
                if (k < cum + hh) { pref |= (unsigned)b << shift; k -= cum; break; }
                cum += hh;
            }
            s_prefix[tid] = pref;
            s_rank[tid]   = k;
        }
        __syncthreads();
    }

    if (tid == 0) {
        const float vlo = u2f(s_prefix[0]);
        const float vhi = u2f(s_prefix[1]);
        const float med = 0.5f * (vlo + vhi);
        out[p] = (med + 48.0f) * (1.0f / 96.0f);
    }
}

// ---------------------------------------------------------------------------
extern "C" void kernel_launch(void* const* d_in, const int* in_sizes, int n_in,
                              void* d_out, int out_size, void* d_ws, size_t ws_size,
                              hipStream_t stream)
{
    (void)in_sizes; (void)n_in; (void)out_size; (void)ws_size;
    const float* pts = (const float*)d_in[0];
    float* out = (float*)d_out;
    float* wsf = (float*)d_ws;

    // workspace layout (8-byte aligned v2f array first):
    v2f*   W     = (v2f*)wsf;                 // 32768 floats (128 KB)
    float* sq    = wsf + 2 * NTILES * 32;     // 8192 floats
    float* nn    = sq + NPTS;                 // 8192 floats
    float* delta = nn + NPTS;                 // 1 float

    prep_kernel<<<NPTS / 256, 256, 0, stream>>>(pts, sq, W);
    nn_kernel<<<NTILES / 8, 256, 0, stream>>>(sq, W, nn);
    reduce_kernel<<<1, 256, 0, stream>>>(nn, delta);
    project_kernel<<<GRID_W * GRID_W, 256, NPTS * 12, stream>>>(pts, delta, out);
}